// CTCLayer_49151605736161
// MI455X (gfx1250) — compile-verified
//
#include <hip/hip_runtime.h>

// CTC forward (keras ctc_batch_cost semantics) for B=64, T=2048, C=128, L=256.
// One block per batch element; one thread per extended-label state (S=513).
// Per-time-step probability row (128 f32 = 512B) staged into LDS with gfx1250
// async DMA (global_load_async_to_lds_b128 / ASYNCcnt), double-buffered and
// issued one full step ahead. Time loop unrolled x2 so all LDS buffer indices
// are compile-time constants; one split-barrier per step.

#define Bb 64
#define Tt 2048
#define Cc 128
#define Ll 256
#define Ss (2 * Ll + 1)       // 513 extended states
#define BLANK (Cc - 1)
#define NEGV (-1.0e30f)
#define EPSV (1e-7f)
#define NTHREADS 544          // 17 waves of 32 (513 active states)

__device__ __forceinline__ float laddexp(float a, float b) {
    // logaddexp(a,b) = max + log1p(exp(min-max)); safe for NEG sentinels
    float mx = fmaxf(a, b);
    float d  = fminf(a, b) - mx;           // <= 0 (0 when a==b, incl. NEG,NEG)
    return mx + log1pf(__expf(d));
}

__device__ __forceinline__ void async_row_b128(unsigned lds_addr, const float* gptr) {
    // per-lane: copy 16 bytes global -> LDS, tracked by ASYNCcnt
    asm volatile("global_load_async_to_lds_b128 %0, %1, off"
                 :: "v"(lds_addr), "v"(gptr) : "memory");
}

__device__ __forceinline__ void wait_async0() {
    asm volatile("s_wait_asynccnt 0" ::: "memory");
}

// One CTC time step: read aOld, write aNew, consume prob row `prow` (in LDS),
// optionally DMA the row for time t+2-ish into LDS at dstLds.
__device__ __forceinline__ void ctc_step(int tid, int cls, bool skip,
                                         const float* __restrict__ aOld,
                                         float* __restrict__ aNew,
                                         const float* __restrict__ prow,
                                         const float* gsrc, unsigned dstLds,
                                         bool doPrefetch)
{
    if (doPrefetch && tid < 32) {
        async_row_b128(dstLds + 16u * (unsigned)tid, gsrc + 4 * tid);
    }
    if (tid < Ss) {
        float aS = aOld[2 + tid];
        float a1 = aOld[1 + tid];
        float a2 = aOld[tid];
        float a  = laddexp(aS, a1);
        a = laddexp(a, skip ? a2 : NEGV);       // branchless: laddexp(x,NEGV)==x
        aNew[2 + tid] = a + __logf(prow[cls] + EPSV);
    }
    if (tid < 32) wait_async0();                 // prefetched row landed
    __syncthreads();                             // aNew + new row visible to all
}

__global__ __launch_bounds__(NTHREADS)
void ctc_fwd_kernel(const float* __restrict__ y_pred,
                    const long long* __restrict__ y_true,
                    float* __restrict__ out)
{
    __shared__ float alphaBuf[2][Ss + 2];   // +2 front pad of NEG: branch-free s-1/s-2
    __shared__ float probs[2][Cc];          // double-buffered per-step prob row

    const int tid = threadIdx.x;
    const int b   = blockIdx.x;

    const float*     yp  = y_pred + (size_t)b * Tt * Cc;
    const long long* lab = y_true + (size_t)b * Ll;

    const unsigned p0 = (unsigned)(size_t)(&probs[0][0]);
    const unsigned p1 = (unsigned)(size_t)(&probs[1][0]);

    // ---- issue async DMA of the t=0 probability row into probs[0] ----
    if (tid < 32) {
        async_row_b128(p0 + 16u * (unsigned)tid, yp + 4 * tid);
    }

    // ---- loop-invariant per-state config (overlaps with DMA) ----
    int  cls  = BLANK;
    bool skip = false;
    if (tid < Ss) {
        if (tid & 1) {
            int li = (tid - 1) >> 1;
            cls  = (int)lab[li];
            skip = (li == 0) ? true : (cls != (int)lab[li - 1]); // s=1: reads NEG pad, harmless
        }
    }
    if (tid < 2) { alphaBuf[0][tid] = NEGV; alphaBuf[1][tid] = NEGV; }

    if (tid < 32) wait_async0();
    __syncthreads();                         // probs[0] (t=0) ready

    // ---- alpha(t=0); prefetch t=1 row underneath ----
    if (tid < 32) {
        async_row_b128(p1 + 16u * (unsigned)tid, yp + Cc + 4 * tid);
    }
    if (tid < Ss) {
        float a0 = NEGV;
        if (tid < 2) a0 = __logf(probs[0][cls] + EPSV);
        alphaBuf[0][2 + tid] = a0;
    }
    if (tid < 32) wait_async0();
    __syncthreads();                         // alpha(0) + probs[1] (t=1) ready

    // ---- serial scan over time, unrolled x2 (2047 steps = 1023 pairs + tail) ----
    const float* gnext = yp + 2 * Cc;        // first row to prefetch is t=2
    #pragma unroll 1
    for (int t = 1; t < Tt - 1; t += 2) {
        // t (odd): consume probs[1], alpha 0 -> 1, prefetch t+1 into probs[0]
        ctc_step(tid, cls, skip, alphaBuf[0], alphaBuf[1], probs[1], gnext, p0, true);
        gnext += Cc;
        // t+1 (even): consume probs[0], alpha 1 -> 0, prefetch t+2 into probs[1]
        ctc_step(tid, cls, skip, alphaBuf[1], alphaBuf[0], probs[0], gnext, p1, true);
        gnext += Cc;
    }
    // tail t = Tt-1 (odd): consume probs[1], alpha 0 -> 1, no prefetch
    ctc_step(tid, cls, skip, alphaBuf[0], alphaBuf[1], probs[1], gnext, p0, false);

    // ---- -log( exp(alpha[S-1]) + exp(alpha[S-2]) ) ----
    if (tid == 0) {
        float ll = laddexp(alphaBuf[1][2 + Ss - 1], alphaBuf[1][2 + Ss - 2]);
        out[b] = -ll;
    }
}

extern "C" void kernel_launch(void* const* d_in, const int* in_sizes, int n_in,
                              void* d_out, int out_size, void* d_ws, size_t ws_size,
                              hipStream_t stream) {
    (void)in_sizes; (void)n_in; (void)out_size; (void)d_ws; (void)ws_size;
    const long long* y_true = (const long long*)d_in[0];   // int64 [64,256]
    const float*     y_pred = (const float*)d_in[1];       // f32   [64,2048,128]
    float*           out    = (float*)d_out;               // f32   [64,1]
    hipLaunchKernelGGL(ctc_fwd_kernel, dim3(Bb), dim3(NTHREADS), 0, stream,
                       y_pred, y_true, out);
}